// Multi_Scale_Region_Distillation_Loss_1_33414845562964
// MI455X (gfx1250) — compile-verified
//
#include <hip/hip_runtime.h>
#include <hip/hip_bf16.h>

typedef __attribute__((ext_vector_type(2))) float v2f;
typedef __attribute__((ext_vector_type(4))) float v4f;
typedef __attribute__((ext_vector_type(8))) float v8f;

#define NSEG 17          // 0..15 real, 16 = dump bucket (weight 0)
#define NUM_OLD 15
#define NUM_CLASS 21

// ---------------------------------------------------------------------------
// Zero the global accumulators (4 scales x 17 segs sums + counts).
// ---------------------------------------------------------------------------
__global__ void init_kernel(float* __restrict__ gsum, int* __restrict__ gcnt) {
    int t = threadIdx.x;
    if (t < 4 * NSEG) { gsum[t] = 0.0f; gcnt[t] = 0; }
}

// ---------------------------------------------------------------------------
// Build the 8x128x128 segment map at full-res sample points (4y,4x).
// pl = (label==0) ? argmax_c(threshold(outputs_old)) : label; seg=min(pl,16).
// Reads only the needed 1/16 of outputs_old rows/cols (8.4 MB, not 128 MB).
// One-shot strided reads -> non-temporal.
// ---------------------------------------------------------------------------
__global__ __launch_bounds__(256) void segmap_kernel(
    const int* __restrict__ labels,      // [8,512,512]
    const float* __restrict__ oo,        // [8,16,512,512]
    signed char* __restrict__ seg)       // [8,128,128]
{
    int tid = blockIdx.x * 256 + threadIdx.x;   // 0..131071
    int b = tid >> 14;
    int r = tid & 16383;
    int y = r >> 7, x = r & 127;
    int fy = y << 2, fx = x << 2;               // full-res coords (y*4, x*4)

    int lbl = __builtin_nontemporal_load(labels + (b << 18) + (fy << 9) + fx);
    int pl = lbl;
    if (lbl == 0) {
        const float* p = oo + (((size_t)b * 16) << 18) + (size_t)((fy << 9) + fx);
        float best = -1.0f;
        int arg = 0;
        #pragma unroll
        for (int c = 0; c < NUM_OLD + 1; ++c) {
            float v = __builtin_nontemporal_load(p + ((size_t)c << 18));
            v = (v < 0.5f) ? 0.0f : v;          // threshold
            if (v > best) { best = v; arg = c; } // strict > => first max (jnp.argmax)
        }
        pl = arg;
    }
    seg[tid] = (signed char)((pl <= NUM_OLD) ? pl : (NUM_OLD + 1));
}

// ---------------------------------------------------------------------------
// Per-scale streaming reduction. Each thread owns 4 contiguous pixels
// (ext-vector float4 -> global_load_b128 NT on both tensors: 480 MB streamed
// once, 2.5x the 192 MB L2, zero reuse), loops over a channel chunk, then
// ds_add_f32 into 17 LDS bins; chunk 0 also counts pixels. The seg map
// (reused by all 4 kernels) stays on the default temporal path.
//   C  = channels, HW = h*w, W = w, K = 128/h (stride into seg map),
//   CH = channels per block (blockIdx.y selects the chunk).
// ---------------------------------------------------------------------------
template<int C, int HW, int W, int K, int CH>
__global__ __launch_bounds__(256) void sqdiff_seg_kernel(
    const float* __restrict__ f, const float* __restrict__ fo,
    const signed char* __restrict__ seg128,
    float* __restrict__ gsum, int* __restrict__ gcnt)
{
    __shared__ float lsum[NSEG];
    __shared__ int   lcnt[NSEG];
    int t = threadIdx.x;
    if (t < NSEG) { lsum[t] = 0.0f; lcnt[t] = 0; }
    __syncthreads();

    int p0 = blockIdx.x * 1024 + t * 4;   // first of 4 contiguous pixels
    int b  = p0 / HW;
    int r  = p0 - b * HW;                 // 4-aligned in-plane offset (one row)
    int y  = r / W;
    int x0 = r - y * W;

    // segment ids (scale grid is a strided subset of the 128x128 seg map)
    int sb = (b << 14) + (y * K << 7) + x0 * K;
    int sA = seg128[sb];
    int sB = seg128[sb + K];
    int sC = seg128[sb + 2 * K];
    int sD = seg128[sb + 3 * K];

    const int c0 = blockIdx.y * CH;
    size_t base = (size_t)(b * C + c0) * HW + (size_t)r;
    float ax = 0.f, ay = 0.f, az = 0.f, aw = 0.f;
    #pragma unroll 4
    for (int c = 0; c < CH; ++c) {
        v4f a = __builtin_nontemporal_load((const v4f*)(f  + base));
        v4f o = __builtin_nontemporal_load((const v4f*)(fo + base));
        float dx = a.x - o.x, dy = a.y - o.y, dz = a.z - o.z, dw = a.w - o.w;
        ax += dx * dx; ay += dy * dy; az += dz * dz; aw += dw * dw;
        base += HW;
    }

    atomicAdd(&lsum[sA], ax);
    atomicAdd(&lsum[sB], ay);
    atomicAdd(&lsum[sC], az);
    atomicAdd(&lsum[sD], aw);
    if (blockIdx.y == 0) {   // count each pixel exactly once
        atomicAdd(&lcnt[sA], 1); atomicAdd(&lcnt[sB], 1);
        atomicAdd(&lcnt[sC], 1); atomicAdd(&lcnt[sD], 1);
    }
    __syncthreads();
    if (t < NSEG) {
        if (lsum[t] != 0.0f) atomicAdd(&gsum[t], lsum[t]);
        if (blockIdx.y == 0 && lcnt[t] != 0) atomicAdd(&gcnt[t], lcnt[t]);
    }
}

// ---------------------------------------------------------------------------
// Final weighted reduction on one wave32 via V_WMMA_F32_16X16X4_F32.
// Seg 16 has weight 0, so the live problem is 16 segs x 4 scales:
//   A[m,k] = scale_w[k] * cls_w[m] * mse[k][m]  (16x4),  B = ones (4x16)
//   D[m,n] = sum_k A[m,k]; column 0 lives in lanes 0 (M=0..7) and 16 (M=8..15).
// ---------------------------------------------------------------------------
__global__ __launch_bounds__(32) void finalize_kernel(
    const float* __restrict__ gsum, const int* __restrict__ gcnt,
    float* __restrict__ out)
{
    __shared__ float red[32];
    int lane = threadIdx.x;
    const float invC[4] = {1.f/256.f, 1.f/512.f, 1.f/1024.f, 1.f/2048.f};
    const float sw[4]   = {1.f, 2.f, 3.f, 4.f};

    int m = lane & 15;
    int kbase = (lane < 16) ? 0 : 2;     // A layout: lanes 0-15 K=0,1; 16-31 K=2,3
    float clsw = (m == 0) ? ((float)NUM_OLD / (float)NUM_CLASS) : 1.0f;

    v2f A;
    #pragma unroll
    for (int j = 0; j < 2; ++j) {
        int k = kbase + j;
        float s = gsum[k * NSEG + m];
        int   c = gcnt[k * NSEG + m];
        float mse = (c > 0) ? (s / (float)c) * invC[k] : 0.0f;
        A[j] = sw[k] * clsw * mse;
    }
    v2f Bm; Bm[0] = 1.0f; Bm[1] = 1.0f;
    v8f Cacc = {};
    v8f D = __builtin_amdgcn_wmma_f32_16x16x4_f32(
        /*neg_a=*/false, A, /*neg_b=*/false, Bm,
        /*c_mod=*/(short)0, Cacc, /*reuse_a=*/false, /*reuse_b=*/false);

    float partial = D[0]+D[1]+D[2]+D[3]+D[4]+D[5]+D[6]+D[7];
    red[lane] = partial;
    __syncthreads();
    if (lane == 0) out[0] = 0.25f * (red[0] + red[16]);   // /len(feats)
}

// ---------------------------------------------------------------------------
extern "C" void kernel_launch(void* const* d_in, const int* in_sizes, int n_in,
                              void* d_out, int out_size, void* d_ws, size_t ws_size,
                              hipStream_t stream) {
    // setup_inputs() dict order:
    const int*   labels = (const int*)  d_in[0];
    const float* oo     = (const float*)d_in[1];
    const float* f0  = (const float*)d_in[2];
    const float* fo0 = (const float*)d_in[3];
    const float* f1  = (const float*)d_in[4];
    const float* fo1 = (const float*)d_in[5];
    const float* f2  = (const float*)d_in[6];
    const float* fo2 = (const float*)d_in[7];
    const float* f3  = (const float*)d_in[8];
    const float* fo3 = (const float*)d_in[9];
    // d_in[10]=num_class(21), d_in[11]=num_old_class(15) — baked in as constants.

    signed char* seg  = (signed char*)d_ws;                       // 128 KB
    float*       gsum = (float*)((char*)d_ws + 131072);           // 4*17 floats
    int*         gcnt = (int*)  ((char*)d_ws + 131072 + 4*NSEG*4);// 4*17 ints

    init_kernel<<<1, 128, 0, stream>>>(gsum, gcnt);
    segmap_kernel<<<512, 256, 0, stream>>>(labels, oo, seg);

    //                    C     HW    W   K  CH        pixblocks, chunks
    sqdiff_seg_kernel< 256, 16384, 128, 1, 32><<<dim3(128,   8), 256, 0, stream>>>(f0, fo0, seg, gsum + 0*NSEG, gcnt + 0*NSEG);
    sqdiff_seg_kernel< 512,  4096,  64, 2, 32><<<dim3( 32,  16), 256, 0, stream>>>(f1, fo1, seg, gsum + 1*NSEG, gcnt + 1*NSEG);
    sqdiff_seg_kernel<1024,  1024,  32, 4, 16><<<dim3(  8,  64), 256, 0, stream>>>(f2, fo2, seg, gsum + 2*NSEG, gcnt + 2*NSEG);
    sqdiff_seg_kernel<2048,   256,  16, 8,  8><<<dim3(  2, 256), 256, 0, stream>>>(f3, fo3, seg, gsum + 3*NSEG, gcnt + 3*NSEG);

    finalize_kernel<<<1, 32, 0, stream>>>(gsum, gcnt, (float*)d_out);
}